// GSHashEncoding_73443940761815
// MI455X (gfx1250) — compile-verified
//
#include <hip/hip_runtime.h>

typedef __attribute__((ext_vector_type(2)))  float v2f;
typedef __attribute__((ext_vector_type(8)))  float v8f;

namespace {
constexpr int kReso  = 2097152;   // points
constexpr int kL0    = 65536;     // level-0 table rows
constexpr int kDim   = 4;         // dims per level
constexpr int kOut   = 48;        // output features
constexpr int kTiles = kReso / 16;
}

// Builtin availability check only makes sense in the device pass; the host
// compiler never has amdgcn builtins.
#if defined(__HIP_DEVICE_COMPILE__) && !__has_builtin(__builtin_amdgcn_wmma_f32_16x16x4_f32)
#error "gfx1250 device toolchain without wmma_f32_16x16x4_f32 builtin"
#endif

// ---------------------------------------------------------------------------
// Exact-f32 WMMA 16x16x4 hash-encoding + linear layer.
//   Per wave-tile of 16 points: feats(16x8) @ W(8x48) as 2 K-chunks x 3 N-tiles
//   = 6 x v_wmma_f32_16x16x4_f32 (vs 192 scalar v_fma).
//
// A layout (32-bit 16x4, ISA 7.12.2): lanes 0-15 hold K=0 (v0), K=1 (v1);
//   lanes 16-31 hold K=2 (v0), K=3 (v1), same M = lane&15.
//   -> lane m gathers dims {0,1}; lane m+16 gathers dims {2,3} of point m,
//      so the 8 gathers per point are split evenly across the wave halves.
//
// Memory policy (roofline: 64 MB map stream + 5.2 MB L2-resident tables in,
// 403 MB write-once stream out; ~20 us at 23.3 TB/s):
//   - maps: non-temporal loads (read exactly once)
//   - code tables: regular loads (keep L2-resident)
//   - output: non-temporal stores (never re-read; don't rinse L2)
// ---------------------------------------------------------------------------
__global__ __launch_bounds__(256) void gs_hash_wmma_f32(
    const float* __restrict__ codes,
    const float* __restrict__ W,
    const int*   __restrict__ map0,
    const int*   __restrict__ map1,
    float*       __restrict__ out)
{
    const int lane = (int)(threadIdx.x & 31);
    const int half = lane >> 4;          // 0 -> dims {0,1}, 1 -> dims {2,3}
    const int m    = lane & 15;          // point-in-tile (A) / column (B, D)
    const int d0   = 2 * half;

    const float* __restrict__ codes1 = codes + (size_t)kL0 * kDim;

    // Loop-invariant B tiles: B[c][t] holds K rows {4c+d0, 4c+d0+1}, col 16t+m.
    // B layout (4x16, KxN): v0: lanes 0-15 K=0 / lanes 16-31 K=2; v1: K=1 / K=3.
    v2f B[2][3];
#pragma unroll
    for (int c = 0; c < 2; ++c)
#pragma unroll
        for (int t = 0; t < 3; ++t) {
            const int n = 16 * t + m;
            B[c][t].x = W[(4 * c + d0    ) * kOut + n];
            B[c][t].y = W[(4 * c + d0 + 1) * kOut + n];
        }

    const int waveId = (int)((blockIdx.x * blockDim.x + threadIdx.x) >> 5);
    const int nWaves = (int)((gridDim.x * blockDim.x) >> 5);

    for (int tile = waveId; tile < kTiles; tile += nWaves) {
        const int p = tile * 16 + m;

        // Two adjacent indices per level for this lane's dims (NT: streamed once).
        const int i00 = __builtin_nontemporal_load(map0 + p * 4 + d0);
        const int i01 = __builtin_nontemporal_load(map0 + p * 4 + d0 + 1);
        const int i10 = __builtin_nontemporal_load(map1 + p * 4 + d0);
        const int i11 = __builtin_nontemporal_load(map1 + p * 4 + d0 + 1);

        // Gather features (tables: 5.2 MB << 192 MB L2 -> stay cached).
        v2f A0, A1;
        A0.x = codes [i00 * 4 + d0];       // feat d0      (K-chunk 0)
        A0.y = codes [i01 * 4 + d0 + 1];   // feat d0+1
        A1.x = codes1[i10 * 4 + d0];       // feat 4+d0    (K-chunk 1)
        A1.y = codes1[i11 * 4 + d0 + 1];   // feat 4+d0+1

        float* __restrict__ o = out + (size_t)tile * 16 * kOut;

#pragma unroll
        for (int t = 0; t < 3; ++t) {
            v8f acc = {};
            acc = __builtin_amdgcn_wmma_f32_16x16x4_f32(
                false, A0, false, B[0][t], (short)0, acc, false, false);
            acc = __builtin_amdgcn_wmma_f32_16x16x4_f32(
                false, A1, false, B[1][t], (short)0, acc, false, false);
            // D layout: vgpr r, lanes 0-15 -> (M=r, N=lane); lanes 16-31 -> (M=r+8).
            // Each NT store covers two contiguous 64B segments of the write stream.
#pragma unroll
            for (int r = 0; r < 8; ++r)
                __builtin_nontemporal_store(
                    acc[r], o + (r + 8 * half) * kOut + 16 * t + m);
        }
    }
}

extern "C" void kernel_launch(void* const* d_in, const int* in_sizes, int n_in,
                              void* d_out, int out_size, void* d_ws, size_t ws_size,
                              hipStream_t stream) {
    const float* codes = (const float*)d_in[0];   // (65536+262144, 4) f32
    const float* W     = (const float*)d_in[1];   // (8, 48) f32
    const int*   map0  = (const int*)  d_in[2];   // (2097152, 4) i32
    const int*   map1  = (const int*)  d_in[3];   // (2097152, 4) i32
    float*       out   = (float*)      d_out;     // (2097152, 48) f32

    // 4096 blocks x 256 threads = 32768 waves; 131072 tiles -> 4 tiles/wave,
    // amortizing the loop-invariant W (B-matrix) register load.
    dim3 grid(4096), block(256);
    gs_hash_wmma_f32<<<grid, block, 0, stream>>>(codes, W, map0, map1, out);
}